// WordEmbeddingModule_43250320671404
// MI455X (gfx1250) — compile-verified
//
#include <hip/hip_runtime.h>
#include <stdint.h>

// Problem constants from the reference
#define B_     32768
#define LMAX_  24
#define M_     10
#define D_     300
#define V_     36
#define F4ROW  (D_ / 4)                  // 75 float4 per output row
#define TOTAL4 (B_ * M_ * F4ROW)         // 24,576,000 float4 outputs

typedef unsigned int v4u  __attribute__((ext_vector_type(4)));
typedef int          v8i_ __attribute__((ext_vector_type(8)));
typedef int          v4i_ __attribute__((ext_vector_type(4)));
typedef float        v4f  __attribute__((ext_vector_type(4)));
typedef float        v2f  __attribute__((ext_vector_type(2)));

__global__ __launch_bounds__(256)
void WordEmbeddingModule_43250320671404_kernel(const float* __restrict__ emb,
                                               const float* __restrict__ pad,
                                               const int*   __restrict__ tokens,
                                               const int*   __restrict__ lengths,
                                               v4f*         __restrict__ out)
{
    // 36 embedding rows + 1 pad row, 300 f32 each = 44,400 bytes of LDS.
    __shared__ __align__(16) float tab[(V_ + 1) * D_];

    // Stage the pad vector into LDS row 36: 150 float2 chunks, one per thread.
    if (threadIdx.x < (D_ / 2))
        ((v2f*)&tab[V_ * D_])[threadIdx.x] = ((const v2f*)pad)[threadIdx.x];

#if __has_builtin(__builtin_amdgcn_tensor_load_to_lds) && \
    __has_builtin(__builtin_amdgcn_s_wait_tensorcnt)
    // TDM: DMA the full 36x300 f32 table global -> LDS with one descriptor.
    if (threadIdx.x < 32) {   // one wave issues; EXEC is ignored by TDM
        uint32_t lds_off = (uint32_t)(uintptr_t)(&tab[0]);   // low 32b = LDS offset
        uint64_t ga      = (uint64_t)(uintptr_t)emb;

        // D# group 0: count=1 | lds_addr | global_addr[56:0] | type=2
        v4u g0;
        g0.x = 1u;
        g0.y = lds_off;
        g0.z = (uint32_t)ga;
        g0.w = ((uint32_t)(ga >> 32) & 0x01FFFFFFu) | 0x80000000u; // addr[56:32] | type=2

        // D# group 1: data_size=4B, tensor_dim0=300, tensor_dim1=36,
        //             tile_dim0=300, tile_dim1=36, tensor_dim0_stride=300
        v8i_ g1;
        g1[0] = (2 << 16);                                    // data_size code 2 = 4B
        g1[1] = (int)((D_ & 0xFFFF) << 16);                   // tensor_dim0[15:0] @ bit48
        g1[2] = (int)((D_ >> 16) | ((V_ & 0xFFFF) << 16));    // dim0 hi / tensor_dim1 lo
        g1[3] = (int)((V_ >> 16) | ((D_ & 0xFFFF) << 16));    // dim1 hi / tile_dim0
        g1[4] = V_;                                           // tile_dim1=36, tile_dim2=0
        g1[5] = D_;                                           // tensor_dim0_stride[31:0]
        g1[6] = 0;
        g1[7] = 0;

        v4i_ gz4 = {0, 0, 0, 0};            // 2-D tensor: groups 2/3 unused
        v8i_ gz8 = {0, 0, 0, 0, 0, 0, 0, 0};
        __builtin_amdgcn_tensor_load_to_lds(g0, g1, gz4, gz4, gz8, 0);
        __builtin_amdgcn_s_wait_tensorcnt(0);
    }
#else
    // Fallback staging if the TDM builtin is unavailable on this toolchain.
    for (int i = threadIdx.x; i < V_ * D_; i += blockDim.x)
        tab[i] = emb[i];
#endif
    __syncthreads();

    const v4f* tab4 = (const v4f*)tab;

    unsigned idx    = blockIdx.x * blockDim.x + threadIdx.x;
    unsigned stride = gridDim.x * blockDim.x;
    for (; idx < (unsigned)TOTAL4; idx += stride) {
        unsigned d4 = idx % F4ROW;          // float4 column within row
        unsigned t  = idx / F4ROW;          // flattened (b, i)
        unsigned i  = t % M_;
        unsigned b  = t / M_;

        const int* tok = tokens + b * LMAX_;
        int L = lengths[b];                 // 1..24, uniform across the 75-lane span
        v4f r;
        if (L < M_) {
            // padding branch: first L rows are token embeddings, rest = pad row (36)
            int row = ((int)i < L) ? tok[i] : V_;
            r = tab4[row * F4ROW + d4];     // ds_load_b128
        } else {
            // bilinear (align_corners) resample along the char axis
            float pos = ((float)i * (float)(L - 1)) / 9.0f;   // i*(L-1)/(M-1)
            int   lo  = (int)floorf(pos);
            int   hi  = lo + 1; if (hi > L - 1) hi = L - 1;
            float w   = pos - (float)lo;
            float iw  = 1.0f - w;
            v4f a = tab4[tok[lo] * F4ROW + d4];
            v4f c = tab4[tok[hi] * F4ROW + d4];
            r = a * iw + c * w;             // matches g_lo*(1-w) + g_hi*w
        }
        // Streaming output: written once, never re-read -> non-temporal b128 store
        __builtin_nontemporal_store(r, out + idx);
    }
}

extern "C" void kernel_launch(void* const* d_in, const int* in_sizes, int n_in,
                              void* d_out, int out_size, void* d_ws, size_t ws_size,
                              hipStream_t stream) {
    const float* emb     = (const float*)d_in[0];   // [36, 300] f32
    const float* pad     = (const float*)d_in[1];   // [1, 300]  f32
    const int*   tokens  = (const int*)d_in[2];     // [B, 24]   i32
    const int*   lengths = (const int*)d_in[3];     // [B]       i32
    v4f*         out     = (v4f*)d_out;             // [B, 10, 300] f32

    (void)in_sizes; (void)n_in; (void)out_size; (void)d_ws; (void)ws_size;

    dim3 grid(4096), block(256);   // grid-stride: ~23 float4 per thread,
                                   // 44 KB LDS staging amortized per block
    hipLaunchKernelGGL(WordEmbeddingModule_43250320671404_kernel,
                       grid, block, 0, stream, emb, pad, tokens, lengths, out);
}